// AdaptiveFilterBank_39865886441550
// MI455X (gfx1250) — compile-verified
//
#include <hip/hip_runtime.h>
#include <cstdint>

#define NF        8
#define FLEN      31
#define PADW      15
#define BATCH     64
#define LEN       131072
#define TILE      4096          // floats per block chunk
#define THREADS   256

typedef float v2f __attribute__((ext_vector_type(2)));
typedef float v8f __attribute__((ext_vector_type(8)));

// ---------------------------------------------------------------------------
// Kernel 1: per-batch effective 31-tap filter via WMMA f32 16x16x4.
//   h^T      = relu(W1 @ features^T + b1)      (D1[n][m], n=filter, m=batch)
//   logits^T = W2 @ h^T + b2                   (D2[n2][m])
// D layout (16x16 f32, 8 VGPRs): VGPR v, lanes0-15 -> M=v, N=lane;
// so lane L<16 holds column m=L for all rows v=0..7  ==> each lane owns one
// batch's full 8-vector => local softmax, then eff[k] = sum_f w_f * fp[f][k].
// One wave (32 threads) handles 16 batches; grid.x = 4.
// ---------------------------------------------------------------------------
__global__ __launch_bounds__(32)
void eff_filter_kernel(const float* __restrict__ features,
                       const float* __restrict__ fp,
                       const float* __restrict__ W1, const float* __restrict__ b1,
                       const float* __restrict__ W2, const float* __restrict__ b2,
                       float* __restrict__ eff_out)
{
    __shared__ float h_lds[16 * 8];          // h[m][f], m=0..15 local batches
    const int lane = threadIdx.x;
    const int m0   = blockIdx.x * 16;
    const int r    = lane & 15;              // A-row / B-col / D-col index
    const int kh   = lane >> 4;              // lane half selects K {0,1} vs {2,3}

    // ---- stage 1: D1 = W1 x features^T  (two K=4 slices) ----
    v8f c1 = {};
    #pragma unroll
    for (int j = 0; j < 2; ++j) {
        const int k0 = 4 * j + 2 * kh;
        const int ra = (r < 8) ? r : 0;      // clamp (rows 8..15 of A are zero)
        const float msk = (r < 8) ? 1.f : 0.f;
        v2f a, b;
        a.x = W1[ra * NF + k0]     * msk;    // A[n][k] = W1[n][k]
        a.y = W1[ra * NF + k0 + 1] * msk;
        b.x = features[(m0 + r) * NF + k0];      // B[k][m] = features[m][k]
        b.y = features[(m0 + r) * NF + k0 + 1];
        c1 = __builtin_amdgcn_wmma_f32_16x16x4_f32(false, a, false, b,
                                                   (short)0, c1, false, false);
    }
    if (lane < 16) {                         // lane L holds h_pre[m0+L][0..7]
        #pragma unroll
        for (int v = 0; v < 8; ++v) {
            float hv = c1[v] + b1[v];
            h_lds[lane * 8 + v] = hv > 0.f ? hv : 0.f;   // relu
        }
    }
    __syncthreads();

    // ---- stage 2: D2 = W2 x h^T ----
    v8f c2 = {};
    #pragma unroll
    for (int j = 0; j < 2; ++j) {
        const int k0 = 4 * j + 2 * kh;
        const int ra = (r < 8) ? r : 0;
        const float msk = (r < 8) ? 1.f : 0.f;
        v2f a, b;
        a.x = W2[ra * NF + k0]     * msk;    // A[n2][k] = W2[n2][k]
        a.y = W2[ra * NF + k0 + 1] * msk;
        b.x = h_lds[r * 8 + k0];             // B[k][m] = h[m][k]
        b.y = h_lds[r * 8 + k0 + 1];
        c2 = __builtin_amdgcn_wmma_f32_16x16x4_f32(false, a, false, b,
                                                   (short)0, c2, false, false);
    }

    // ---- per-lane softmax + filter combine ----
    if (lane < 16) {
        float lg[NF];
        float mx = -3.4e38f;
        #pragma unroll
        for (int f = 0; f < NF; ++f) { lg[f] = c2[f] + b2[f]; mx = fmaxf(mx, lg[f]); }
        float s = 0.f;
        #pragma unroll
        for (int f = 0; f < NF; ++f) { lg[f] = __expf(lg[f] - mx); s += lg[f]; }
        const float inv = 1.f / s;
        const int bb = m0 + lane;
        #pragma unroll
        for (int k = 0; k < FLEN; ++k) {
            float e = 0.f;
            #pragma unroll
            for (int f = 0; f < NF; ++f)
                e = fmaf(lg[f] * inv, fp[f * FLEN + k], e);
            eff_out[bb * FLEN + k] = e;
        }
    }
}

// ---------------------------------------------------------------------------
// Kernel 2: HBM-bound depthwise conv, out[b,l] = sum_k eff[b,k]*x[b,l+k-15].
// Stage TILE+halo into LDS with CDNA5 async loads (ASYNCcnt), then each
// thread produces 16 contiguous outputs from a 48-float register window.
// Exactly-once HBM read of x and write of out.
// ---------------------------------------------------------------------------
__global__ __launch_bounds__(THREADS)
void conv_kernel(const float* __restrict__ x,
                 const float* __restrict__ eff,
                 float* __restrict__ out)
{
    __shared__ float smem[TILE + 32];        // [l0-16 .. l0+TILE+16)
    const int t     = threadIdx.x;
    const int chunk = blockIdx.x;
    const int b     = blockIdx.y;
    const long row  = (long)b * LEN;
    const int  l0   = chunk * TILE;
    const float* xrow = x + row;

    // async copy global -> LDS, 16B per lane per issue (1032 float4s total)
    for (int i = t; i < (TILE + 32) / 4; i += THREADS) {
        int g = l0 - 16 + i * 4;                       // element index in row
        g = g < 0 ? 0 : (g > LEN - 4 ? LEN - 4 : g);   // clamp (fixed up below)
        uint64_t ga = (uint64_t)(uintptr_t)(xrow + g);
        uint32_t la = (uint32_t)(uintptr_t)(&smem[i * 4]);
        asm volatile("global_load_async_to_lds_b128 %0, %1, off"
                     :: "v"(la), "v"(ga) : "memory");
    }
    asm volatile("s_wait_asynccnt 0x0" ::: "memory");
    // zero-pad the out-of-row halo
    if (l0 == 0 && t < 16)          smem[t] = 0.f;
    if (l0 + TILE == LEN && t < 16) smem[TILE + 16 + t] = 0.f;
    __syncthreads();

    // broadcast 31-tap filter for this batch
    float effr[FLEN];
    const float* effp = eff + b * FLEN;
    #pragma unroll
    for (int k = 0; k < FLEN; ++k) effr[k] = effp[k];

    // window: outputs p = pb..pb+15 need smem[pb+1 .. pb+46]; load 48 aligned
    const int pb = t * 16;
    float w[48];
    const float4* sv = (const float4*)(&smem[pb]);
    #pragma unroll
    for (int i = 0; i < 12; ++i) {
        float4 v = sv[i];
        w[4 * i + 0] = v.x; w[4 * i + 1] = v.y;
        w[4 * i + 2] = v.z; w[4 * i + 3] = v.w;
    }

    float acc[16];
    #pragma unroll
    for (int j = 0; j < 16; ++j) acc[j] = 0.f;
    #pragma unroll
    for (int k = 0; k < FLEN; ++k)
        #pragma unroll
        for (int j = 0; j < 16; ++j)
            acc[j] = fmaf(effr[k], w[1 + j + k], acc[j]);

    float* orow = out + row + l0 + pb;
    #pragma unroll
    for (int j4 = 0; j4 < 4; ++j4) {
        float4 o = { acc[4 * j4 + 0], acc[4 * j4 + 1],
                     acc[4 * j4 + 2], acc[4 * j4 + 3] };
        ((float4*)orow)[j4] = o;
    }
}

// ---------------------------------------------------------------------------
extern "C" void kernel_launch(void* const* d_in, const int* in_sizes, int n_in,
                              void* d_out, int out_size, void* d_ws, size_t ws_size,
                              hipStream_t stream)
{
    const float* x        = (const float*)d_in[0];
    const float* features = (const float*)d_in[1];
    const float* fp       = (const float*)d_in[2];
    const float* W1       = (const float*)d_in[3];
    const float* b1       = (const float*)d_in[4];
    const float* W2       = (const float*)d_in[5];
    const float* b2       = (const float*)d_in[6];
    float* out = (float*)d_out;
    float* eff = (float*)d_ws;              // 64*31 floats of scratch

    eff_filter_kernel<<<dim3(BATCH / 16), dim3(32), 0, stream>>>(
        features, fp, W1, b1, W2, b2, eff);
    conv_kernel<<<dim3(LEN / TILE, BATCH), dim3(THREADS), 0, stream>>>(
        x, eff, out);
}